// LSTM_54786602828143
// MI455X (gfx1250) — compile-verified
//
#include <hip/hip_runtime.h>
#include <hip/hip_bf16.h>

typedef __attribute__((ext_vector_type(2))) float v2f;
typedef __attribute__((ext_vector_type(8))) float v8f;

#define B_   64
#define T_   512
#define F_   512
#define H_   1024
#define G4_  4096
#define EPS_ 1e-5f
#define NBLK 128

__device__ __forceinline__ float sigmf(float x) {
    return 1.0f / (1.0f + __expf(-x));
}

__global__ void lstm_zero_ws(float* __restrict__ p, int n) {
    int i = blockIdx.x * blockDim.x + threadIdx.x;
    if (i < n) p[i] = 0.0f;
}

// Persistent LSTM: one kernel runs all T=512 steps.
// Each step: g = [x_t ; h_prev] @ [W_XH ; W_HH] + bias, gate math + per-step BN.
// Block owns (batch half rh: 32 rows) x (hidden units [j0, j0+16)) -> the cell
// update is block-local; only h crosses blocks (device-scope barrier).
// grid = 128 blocks, block = 128 threads = 4 wave32 (1 wave/SIMD on >=128 WGPs).
// Each wave: one 16-row tile x two gate tiles (A fragment amortized over 2 WMMAs).
__global__ __launch_bounds__(128) void lstm_persist(
    const float* __restrict__ inputs,   // [B, T, F]
    const float* __restrict__ W_XH,     // [F, 4H]
    const float* __restrict__ W_HH,     // [H, 4H]
    const float* __restrict__ bias,     // [4H]
    const float* __restrict__ scale,    // [T, H]
    const float* __restrict__ offset,   // [T, H]
    const float* __restrict__ pop_mean, // [T, H]
    const float* __restrict__ pop_var,  // [T, H]
    float* __restrict__ h0,             // [B, H] ping
    float* __restrict__ h1,             // [B, H] pong
    float* __restrict__ c,              // [B, H], in place (block-private slice)
    unsigned* __restrict__ bar,         // global barrier counter (zeroed)
    float* __restrict__ out)            // [B, H] final hidden state
{
    __shared__ float gsh[32][68];   // [local row][gate*16 + jj], padded stride

    const int tid  = threadIdx.x;
    const int lane = tid & 31;
    const int wid  = tid >> 5;            // 0..3
    const int r    = wid & 1;             // row tile within this block's half
    const int gA   = (wid >> 1) * 2;      // this wave handles gates gA and gA+1
    const int rh   = blockIdx.x & 1;      // batch half: rows rh*32 .. rh*32+31
    const int j0   = (blockIdx.x >> 1) * 16;  // hidden-unit base for this block
    const int l15  = lane & 15;
    const int half = lane >> 4;           // 0: K={k,k+1}, 1: K={k+2,k+3}
    const int koff = half * 2;
    const int mrow = rh * 32 + r * 16 + l15;  // batch row (M)

    // loop-invariant bases, hoisted out of the recurrence
    const float* arow_base = inputs + (size_t)mrow * T_ * F_ + koff;   // += F_ per t
    const float* wpX_base  = W_XH + (size_t)koff * G4_ + gA * H_ + j0 + l15;
    const float* wpH_base  = W_HH + (size_t)koff * G4_ + gA * H_ + j0 + l15;
    const size_t hrow_off  = (size_t)mrow * H_ + koff;

    for (int t = 0; t < T_; ++t) {
        const float* hin  = (t & 1) ? h1 : h0;
        float*       hout = (t & 1) ? h0 : h1;

        v8f accA = {};  // gate gA   tile, 16x16 f32 accumulator
        v8f accB = {};  // gate gA+1 tile

        // ---------------- phase 1: x_t @ W_XH  (K = 512) ----------------
        {
            const float* arow = arow_base + (size_t)t * F_;
            const float* wp   = wpX_base;
            #pragma unroll 4
            for (int kk = 0; kk < F_; kk += 4) {
                v2f a;  a.x  = arow[0];        a.y  = arow[1];
                v2f ba; ba.x = wp[0];          ba.y = wp[G4_];
                v2f bb; bb.x = wp[H_];         bb.y = wp[H_ + G4_];
                accA = __builtin_amdgcn_wmma_f32_16x16x4_f32(
                           false, a, false, ba, (short)0, accA, false, false);
                accB = __builtin_amdgcn_wmma_f32_16x16x4_f32(
                           false, a, false, bb, (short)0, accB, false, false);
                arow += 4;
                wp   += 4 * G4_;
            }
        }

        // ---------------- phase 2: h_prev @ W_HH  (K = 1024) ----------------
        {
            const float* hrow = hin + hrow_off;
            const float* wp   = wpH_base;
            #pragma unroll 4
            for (int kk = 0; kk < H_; kk += 4) {
                v2f a;  a.x  = hrow[0];        a.y  = hrow[1];
                v2f ba; ba.x = wp[0];          ba.y = wp[G4_];
                v2f bb; bb.x = wp[H_];         bb.y = wp[H_ + G4_];
                accA = __builtin_amdgcn_wmma_f32_16x16x4_f32(
                           false, a, false, ba, (short)0, accA, false, false);
                accB = __builtin_amdgcn_wmma_f32_16x16x4_f32(
                           false, a, false, bb, (short)0, accB, false, false);
                hrow += 4;
                wp   += 4 * G4_;
            }
        }

        // C/D layout: VGPR v, lanes 0-15 -> M = v, N = lane; lanes 16-31 -> M = v+8
        #pragma unroll
        for (int v = 0; v < 8; ++v) {
            int row = r * 16 + v + 8 * half;      // local row 0..31
            gsh[row][gA * 16 + l15]       = accA[v];
            gsh[row][(gA + 1) * 16 + l15] = accB[v];
        }
        __syncthreads();

        // ---------------- elementwise: gates, cell update, per-step BN ----------------
        #pragma unroll
        for (int it = 0; it < 4; ++it) {
            int idx  = tid + 128 * it;    // 0..511 over 32 rows x 16 cols
            int bloc = idx >> 4;
            int jj   = idx & 15;
            int b    = rh * 32 + bloc;
            int j    = j0 + jj;

            float f_ = gsh[bloc][jj]      + bias[j];
            float i_ = gsh[bloc][16 + jj] + bias[H_ + j];
            float o_ = gsh[bloc][32 + jj] + bias[2 * H_ + j];
            float gb = gsh[bloc][48 + jj] + bias[3 * H_ + j];

            size_t cj   = (size_t)b * H_ + j;
            float cold  = c[cj];
            float cf    = sigmf(f_ + 1.0f) * cold + sigmf(i_) * tanhf(gb);
            c[cj] = cf;

            size_t pj  = (size_t)t * H_ + j;
            float iss  = scale[pj] * rsqrtf(pop_var[pj] + EPS_);
            float cn   = (cf - pop_mean[pj]) * iss + offset[pj];
            float hv   = sigmf(o_) * tanhf(cn);
            hout[cj] = hv;
            if (t == T_ - 1) out[cj] = hv;
        }

        if (t == T_ - 1) break;   // no barrier needed after the last step

        // ---------------- device-scope barrier between timesteps ----------------
        if (tid == 0) {
            __threadfence();                         // release h stores (global_wb)
            atomicAdd(bar, 1u);
            const unsigned target = (unsigned)NBLK * (unsigned)(t + 1);
            while (__hip_atomic_load(bar, __ATOMIC_ACQUIRE,
                                     __HIP_MEMORY_SCOPE_AGENT) < target) {
                __builtin_amdgcn_s_sleep(8);         // keep polling wave off issue port
            }
        }
        __syncthreads();   // also fences gsh reuse for the next iteration
    }
}

extern "C" void kernel_launch(void* const* d_in, const int* in_sizes, int n_in,
                              void* d_out, int out_size, void* d_ws, size_t ws_size,
                              hipStream_t stream) {
    const float* inputs   = (const float*)d_in[0];
    const float* W_XH     = (const float*)d_in[1];
    const float* W_HH     = (const float*)d_in[2];
    const float* bias     = (const float*)d_in[3];
    const float* scale    = (const float*)d_in[4];
    const float* offset   = (const float*)d_in[5];
    const float* pop_mean = (const float*)d_in[6];
    const float* pop_var  = (const float*)d_in[7];

    float* ws = (float*)d_ws;
    const int HB = B_ * H_;                 // 65536 floats
    float* h0 = ws;
    float* h1 = ws + HB;
    float* cc = ws + 2 * HB;
    unsigned* bar = (unsigned*)(ws + 3 * HB);

    // h0, c, and the barrier counter must start at zero (d_ws is poisoned)
    lstm_zero_ws<<<(3 * HB + 16 + 255) / 256, 256, 0, stream>>>(ws, 3 * HB + 16);

    lstm_persist<<<NBLK, 128, 0, stream>>>(
        inputs, W_XH, W_HH, bias, scale, offset, pop_mean, pop_var,
        h0, h1, cc, bar, (float*)d_out);
}